// ExplainerCompatibleGinGru_80908593922430
// MI455X (gfx1250) — compile-verified
//
#include <hip/hip_runtime.h>
#include <hip/hip_bf16.h>

typedef __attribute__((ext_vector_type(2))) float v2f;
typedef __attribute__((ext_vector_type(8))) float v8f;

#define EMB     32
#define GIN_H   512
#define HGRU    512
#define NNODE   36
#define GB      64      // GIN batch (2*B)
#define BB      32      // B
#define TMAX    37
#define NEDGE   512

// ---------------------------------------------------------------------------
// gfx1250 async global->LDS copy (GVS mode: saddr base + 32-bit voffset),
// tracked by ASYNCcnt.  LDS byte address comes from the low 32 bits of the
// generic shared-aperture pointer (flat->LDS mapping truncates to [31:0]).
// ---------------------------------------------------------------------------
__device__ __forceinline__ void async_copy_b128(unsigned lds_addr, unsigned voff,
                                                unsigned long long sbase) {
  asm volatile("global_load_async_to_lds_b128 %0, %1, %2"
               :: "v"(lds_addr), "v"(voff), "s"(sbase) : "memory");
}
__device__ __forceinline__ void wait_asynccnt0() {
  asm volatile("s_wait_asynccnt 0x0" ::: "memory");
}

// ---------------------------------------------------------------------------
// WMMA fp32 GEMM: C(MxN) = A(MxK) * B (+bias, optional ReLU).
// TRANSB=0: B stored (K,N) row-major.  TRANSB=1: B stored (N,K) row-major (@W.T).
// One wave32 computes one 16x16 C tile via V_WMMA_F32_16X16X4_F32, 8 waves/block.
// ---------------------------------------------------------------------------
template<int TRANSB, int RELU>
__global__ __launch_bounds__(256)
void wmma_gemm_bias(const float* __restrict__ A, const float* __restrict__ Bm,
                    const float* __restrict__ bias, float* __restrict__ C,
                    int M, int N, int K) {
  const int ntN   = N >> 4;
  const int tiles = (M >> 4) * ntN;
  int tile = blockIdx.x * 8 + (threadIdx.x >> 5);
  if (tile >= tiles) return;                 // whole wave exits together
  const int lane = threadIdx.x & 31;
  const int half = lane >> 4;
  const int l    = lane & 15;
  const int tm = (tile / ntN) << 4;
  const int tn = (tile % ntN) << 4;
  const float* arow = A + (size_t)(tm + l) * K;
  const int n = tn + l;
  v8f c = {0.f,0.f,0.f,0.f,0.f,0.f,0.f,0.f};
  for (int k0 = 0; k0 < K; k0 += 4) {
    const int ka = k0 + 2 * half;
    v2f a, b;
    a.x = arow[ka];
    a.y = arow[ka + 1];
    if (TRANSB) {
      const float* bc = Bm + (size_t)n * K + ka;
      b.x = bc[0];
      b.y = bc[1];
    } else {
      b.x = Bm[(size_t)ka * N + n];
      b.y = Bm[(size_t)(ka + 1) * N + n];
    }
    c = __builtin_amdgcn_wmma_f32_16x16x4_f32(false, a, false, b, (short)0, c,
                                              false, false);
  }
  const float bn = bias ? bias[n] : 0.f;
  const int mb = tm + 8 * half;
#pragma unroll
  for (int r = 0; r < 8; ++r) {
    float v = c[r] + bn;
    if (RELU) v = fmaxf(v, 0.f);
    C[(size_t)(mb + r) * N + n] = v;
  }
}

// ---------------------------------------------------------------------------
// Fused GRU step: one launch per time step (critical serial path).
// Grid = 32 blocks; block owns state columns j0..j0+15 and computes the three
// gate projections gr[:, {j0, 512+j0, 1024+j0}+0..15] = h_in @ Whh.T + bhh as
// six 16x16 WMMA tiles (M=32 split in two).  Whh rows are staged into LDS with
// double-buffered global_load_async_to_lds_b128 (ASYNCcnt), then the gate math
// updates h_out for the owned columns (h double-buffered across steps).
// ---------------------------------------------------------------------------
#define KC   64                      // K chunk
#define NCHK (HGRU / KC)             // 8 chunks
#define BPITCH 68                    // LDS row pitch (floats): 68 % 64 = 4 -> no bank conflicts

__global__ __launch_bounds__(256)
void gru_step_fused(const float* __restrict__ h_in, float* __restrict__ h_out,
                    const float* __restrict__ Whh, const float* __restrict__ bhh,
                    const float* __restrict__ GIt, const int* __restrict__ LOS,
                    int t) {
  __shared__ float sB[2][48][BPITCH];   // staged Whh chunk: 3 groups x 16 rows
  __shared__ float sC[6][16][17];       // gr tiles (+bhh)

  const int tid  = threadIdx.x;
  const int j0   = blockIdx.x << 4;     // owned column slice
  const int w    = tid >> 5;            // wave id 0..7
  const int lane = tid & 31;
  const int half = lane >> 4;
  const int l    = lane & 15;
  const unsigned long long whh_base = (unsigned long long)(uintptr_t)Whh;

  // prefetch of one K-chunk: 48 rows x 64 floats = 768 x b128, 3 per thread
  auto prefetch = [&](int bi, int kc0) {
#pragma unroll
    for (int i = 0; i < 3; ++i) {
      int idx = tid + (i << 8);         // 0..767
      int row = idx >> 4;               // 0..47
      int seg = idx & 15;               // 16B segment
      int grow = ((row >> 4) << 9) + j0 + (row & 15);   // (row/16)*512 + j0 + row%16
      unsigned voff = (unsigned)(((grow << 9) + kc0 + (seg << 2)) << 2); // bytes
      unsigned laddr = (unsigned)(uintptr_t)&sB[bi][row][seg << 2];
      async_copy_b128(laddr, voff, whh_base);
    }
  };

  // tile assignment: waves 0..5 -> (g = w/2 in 0..2, tm = (w&1)*16)
  const int g  = w >> 1;
  const int tm = (w & 1) << 4;
  const float* arow = h_in + (size_t)(tm + l) * HGRU;

  v8f c = {0.f,0.f,0.f,0.f,0.f,0.f,0.f,0.f};

  prefetch(0, 0);
  for (int ch = 0; ch < NCHK; ++ch) {
    wait_asynccnt0();
    __syncthreads();                       // chunk ch resident in sB[ch&1]
    if (ch + 1 < NCHK) prefetch((ch + 1) & 1, (ch + 1) * KC);
    if (w < 6) {
      const int bi = ch & 1;
      const int k0 = ch * KC;
      const float* brow = &sB[bi][(g << 4) + l][0];
#pragma unroll
      for (int kl = 0; kl < KC; kl += 4) {
        const int ka = kl + 2 * half;
        v2f a, b;
        a.x = arow[k0 + ka];
        a.y = arow[k0 + ka + 1];
        b.x = brow[ka];
        b.y = brow[ka + 1];
        c = __builtin_amdgcn_wmma_f32_16x16x4_f32(false, a, false, b, (short)0,
                                                  c, false, false);
      }
    }
    __syncthreads();                       // all reads done before overwrite
  }

  // stash gr tiles (+ bhh) in LDS
  if (w < 6) {
    const float bn = bhh[(g << 9) + j0 + l];
#pragma unroll
    for (int r = 0; r < 8; ++r)
      sC[w][(half << 3) + r][l] = c[r] + bn;
  }
  __syncthreads();

  // gate update for owned columns (512 elements, 2 per thread)
#pragma unroll
  for (int i = 0; i < 2; ++i) {
    int e  = tid + (i << 8);               // 0..511
    int b  = e >> 4;                       // 0..31
    int jl = e & 15;
    int hb = b >> 4;                       // M half
    int br = b & 15;                       // row in tile
    int j  = j0 + jl;
    float hr = sC[hb][br][jl];
    float hz = sC[2 + hb][br][jl];
    float hn = sC[4 + hb][br][jl];
    const float* gi = GIt + (size_t)b * (3 * HGRU) + j;
    float hold = h_in[(size_t)b * HGRU + j];
    float r = 1.f / (1.f + __expf(-(gi[0] + hr)));
    float z = 1.f / (1.f + __expf(-(gi[HGRU] + hz)));
    float n = tanhf(gi[2 * HGRU] + r * hn);
    float hnew = (1.f - z) * n + z * hold;
    h_out[(size_t)b * HGRU + j] = (t < LOS[b]) ? hnew : hold;
  }
}

// ---------------------------------------------------------------------------
// Build h0 (2304 x 32): row b*36+n ; b==0 -> x_emb[0:36], b==32 -> x_emb[36:72]
// ---------------------------------------------------------------------------
__global__ void build_h0(const float* __restrict__ xemb, float* __restrict__ h0) {
  int row = blockIdx.x;
  int b = row / NNODE, n = row % NNODE;
  int c = threadIdx.x;                  // 32 threads
  float v = 0.f;
  if (b == 0)        v = xemb[n * EMB + c];
  else if (b == BB)  v = xemb[(NNODE + n) * EMB + c];
  h0[(size_t)row * EMB + c] = v;
}

// xin[b,n,:] = h[b,n,:] + sum_{e: dst[e]==n} h[b,src[e],:]  (deterministic gather)
__global__ __launch_bounds__(256)
void gin_aggregate(const float* __restrict__ h, const int* __restrict__ edge,
                   float* __restrict__ xin, int W) {
  __shared__ int s_src[NEDGE];
  __shared__ int s_dst[NEDGE];
  int row = blockIdx.x;
  int b = row / NNODE, n = row % NNODE;
  for (int e = threadIdx.x; e < NEDGE; e += blockDim.x) {
    s_src[e] = edge[e];
    s_dst[e] = edge[NEDGE + e];
  }
  __syncthreads();
  const float* hb = h + (size_t)b * NNODE * W;
  for (int c = threadIdx.x; c < W; c += blockDim.x) {
    float acc = hb[(size_t)n * W + c];
    for (int e = 0; e < NEDGE; ++e)
      if (s_dst[e] == n) acc += hb[(size_t)s_src[e] * W + c];
    xin[(size_t)row * W + c] = acc;
  }
}

// R = relu(layernorm(Y) * g + be)
__global__ __launch_bounds__(256)
void ln_relu(const float* __restrict__ Y, const float* __restrict__ g,
             const float* __restrict__ be, float* __restrict__ R) {
  const int N = GIN_H;
  __shared__ float s1[256];
  __shared__ float s2[256];
  int row = blockIdx.x;
  const float* y = Y + (size_t)row * N;
  float s = 0.f, ss = 0.f;
  for (int c = threadIdx.x; c < N; c += blockDim.x) {
    float v = y[c];
    s += v; ss += v * v;
  }
  s1[threadIdx.x] = s; s2[threadIdx.x] = ss;
  __syncthreads();
  for (int off = 128; off > 0; off >>= 1) {
    if ((int)threadIdx.x < off) {
      s1[threadIdx.x] += s1[threadIdx.x + off];
      s2[threadIdx.x] += s2[threadIdx.x + off];
    }
    __syncthreads();
  }
  float mean = s1[0] / N;
  float var  = s2[0] / N - mean * mean;
  float inv  = rsqrtf(var + 1e-5f);
  for (int c = threadIdx.x; c < N; c += blockDim.x) {
    float v = (y[c] - mean) * inv * g[c] + be[c];
    R[(size_t)row * N + c] = fmaxf(v, 0.f);
  }
}

// gres[b, layer*512 + c] = sum_n h[b,n,c]
__global__ __launch_bounds__(256)
void pool_sum(const float* __restrict__ h, float* __restrict__ gres, int layer) {
  int b = blockIdx.x;
  for (int c = threadIdx.x; c < GIN_H; c += blockDim.x) {
    float acc = 0.f;
    for (int n = 0; n < NNODE; ++n)
      acc += h[((size_t)b * NNODE + n) * GIN_H + c];
    gres[(size_t)b * (3 * GIN_H) + (size_t)layer * GIN_H + c] = acc;
  }
}

// padded[t,b,:] = ad[b] if t < LOS[b]-1 ; dis[b] if t == LOS[b]-1 ; else 0
__global__ __launch_bounds__(256)
void build_padded(const float* __restrict__ gres, const int* __restrict__ LOS,
                  float* __restrict__ padded) {
  int b = blockIdx.x, t = blockIdx.y;
  int los = LOS[b];
  const float* ad = gres + (size_t)b * (3 * GIN_H);
  const float* di = gres + (size_t)(b + BB) * (3 * GIN_H);
  float* o = padded + ((size_t)t * BB + b) * (3 * GIN_H);
  for (int c = threadIdx.x; c < 3 * GIN_H; c += blockDim.x) {
    float v = 0.f;
    if (t < los - 1)       v = ad[c];
    else if (t == los - 1) v = di[c];
    o[c] = v;
  }
}

__global__ void zero_fill(float* __restrict__ p, int n) {
  int i = blockIdx.x * blockDim.x + threadIdx.x;
  if (i < n) p[i] = 0.f;
}

// out[0] = hid[0,:] . Wc2 + bc2
__global__ __launch_bounds__(256)
void final_out(const float* __restrict__ hid, const float* __restrict__ Wc2,
               const float* __restrict__ bc2, float* __restrict__ out) {
  __shared__ float s[256];
  float acc = 0.f;
  for (int c = threadIdx.x; c < 2 * HGRU; c += blockDim.x)
    acc += hid[c] * Wc2[c];
  s[threadIdx.x] = acc;
  __syncthreads();
  for (int off = 128; off > 0; off >>= 1) {
    if ((int)threadIdx.x < off) s[threadIdx.x] += s[threadIdx.x + off];
    __syncthreads();
  }
  if (threadIdx.x == 0) out[0] = s[0] + bc2[0];
}

static inline dim3 gemm_grid(int M, int N) {
  int t = (M / 16) * (N / 16);
  return dim3((unsigned)((t + 7) / 8));
}

extern "C" void kernel_launch(void* const* d_in, const int* in_sizes, int n_in,
                              void* d_out, int out_size, void* d_ws, size_t ws_size,
                              hipStream_t stream) {
  (void)in_sizes; (void)n_in; (void)out_size; (void)ws_size;
  const float* xemb = (const float*)d_in[0];
  const int*   edge = (const int*)d_in[1];
  const int*   LOS  = (const int*)d_in[2];
  const float* W1a  = (const float*)d_in[3];
  const float* b1a  = (const float*)d_in[4];
  const float* g1   = (const float*)d_in[5];
  const float* be1  = (const float*)d_in[6];
  const float* W1b  = (const float*)d_in[7];
  const float* b1b  = (const float*)d_in[8];
  const float* Wha  = (const float*)d_in[9];
  const float* bha  = (const float*)d_in[10];
  const float* gh   = (const float*)d_in[11];
  const float* beh  = (const float*)d_in[12];
  const float* Whb  = (const float*)d_in[13];
  const float* bhb  = (const float*)d_in[14];
  const float* Wih  = (const float*)d_in[15];
  const float* Whh  = (const float*)d_in[16];
  const float* bih  = (const float*)d_in[17];
  const float* bhh  = (const float*)d_in[18];
  const float* Wc1  = (const float*)d_in[19];
  const float* bc1  = (const float*)d_in[20];
  const float* Wc2  = (const float*)d_in[21];
  const float* bc2  = (const float*)d_in[22];

  const int ROWS = GB * NNODE;           // 2304
  float* ws = (float*)d_ws;
  float* hbuf   = ws;  ws += (size_t)ROWS * GIN_H;            // h (layer output)
  float* xin    = ws;  ws += (size_t)ROWS * GIN_H;            // h+agg / LN output (reused)
  float* ybuf   = ws;  ws += (size_t)ROWS * GIN_H;            // pre-LN GEMM output
  float* gres   = ws;  ws += (size_t)GB * 3 * GIN_H;          // pooled concat (64,1536)
  float* padded = ws;  ws += (size_t)TMAX * BB * 3 * GIN_H;   // GRU inputs (37,32,1536)
  float* GI     = ws;  ws += (size_t)TMAX * BB * 3 * HGRU;    // hoisted x@Wih.T+bih
  float* hA     = ws;  ws += (size_t)BB * HGRU;               // GRU state ping
  float* hB     = ws;  ws += (size_t)BB * HGRU;               // GRU state pong
  float* hid    = ws;  ws += (size_t)BB * 2 * HGRU;           // classifier hidden

  // ---- GIN layer 1 (K = 32) ----
  build_h0<<<ROWS, EMB, 0, stream>>>(xemb, hbuf);
  gin_aggregate<<<ROWS, 256, 0, stream>>>(hbuf, edge, xin, EMB);
  wmma_gemm_bias<0,0><<<gemm_grid(ROWS, GIN_H), 256, 0, stream>>>(xin, W1a, b1a, ybuf, ROWS, GIN_H, EMB);
  ln_relu<<<ROWS, 256, 0, stream>>>(ybuf, g1, be1, xin);
  wmma_gemm_bias<0,0><<<gemm_grid(ROWS, GIN_H), 256, 0, stream>>>(xin, W1b, b1b, hbuf, ROWS, GIN_H, GIN_H);
  pool_sum<<<GB, 256, 0, stream>>>(hbuf, gres, 0);

  // ---- GIN layers 2,3 (K = 512, shared weights) ----
  for (int l = 1; l < 3; ++l) {
    gin_aggregate<<<ROWS, 256, 0, stream>>>(hbuf, edge, xin, GIN_H);
    wmma_gemm_bias<0,0><<<gemm_grid(ROWS, GIN_H), 256, 0, stream>>>(xin, Wha, bha, ybuf, ROWS, GIN_H, GIN_H);
    ln_relu<<<ROWS, 256, 0, stream>>>(ybuf, gh, beh, xin);
    wmma_gemm_bias<0,0><<<gemm_grid(ROWS, GIN_H), 256, 0, stream>>>(xin, Whb, bhb, hbuf, ROWS, GIN_H, GIN_H);
    pool_sum<<<GB, 256, 0, stream>>>(hbuf, gres, l);
  }

  // ---- GRU: hoist input projection out of the scan ----
  build_padded<<<dim3(BB, TMAX), 256, 0, stream>>>(gres, LOS, padded);
  wmma_gemm_bias<1,0><<<gemm_grid(TMAX * BB, 3 * HGRU), 256, 0, stream>>>(
      padded, Wih, bih, GI, TMAX * BB, 3 * HGRU, 3 * GIN_H);

  zero_fill<<<(BB * HGRU + 255) / 256, 256, 0, stream>>>(hA, BB * HGRU);

  // ---- GRU scan: one fused kernel per step (double-buffered state) ----
  float* hcur = hA;
  float* hnxt = hB;
  for (int t = 0; t < TMAX; ++t) {
    gru_step_fused<<<BB, 256, 0, stream>>>(
        hcur, hnxt, Whh, bhh, GI + (size_t)t * BB * 3 * HGRU, LOS, t);
    float* tmp = hcur; hcur = hnxt; hnxt = tmp;
  }

  // ---- Classifier ----
  wmma_gemm_bias<0,1><<<gemm_grid(BB, 2 * HGRU), 256, 0, stream>>>(
      hcur, Wc1, bc1, hid, BB, 2 * HGRU, HGRU);
  final_out<<<1, 256, 0, stream>>>(hid, Wc2, bc2, (float*)d_out);
}